// Encoder_44633300140730
// MI455X (gfx1250) — compile-verified
//
#include <hip/hip_runtime.h>

typedef __attribute__((ext_vector_type(2))) float v2f;
typedef __attribute__((ext_vector_type(8))) float v8f;

#define HW     240
#define NPIX   57600
#define BATCH  8
#define KF     512
#define KP     256
#define KG     128
#define NA     4
#define FD     128   // FDIM

// flat float offsets into d_out (return-tuple order)
#define OFF_FRONT  0ul
#define OFF_PIDX   8192ul
#define OFF_GIDX   12288ul
#define OFF_DIST   14336ul
#define OFF_FENC   4208640ul
#define OFF_AENC   5257216ul
#define OFF_PENC   5265408ul
#define OFF_GENC   5789696ul

// ---------------------------------------------------------------------------
// Kernel 1: ordered nonzero compaction (row-major, matches jnp.nonzero).
// One wave per (batch, array). 24 waves total.
// ---------------------------------------------------------------------------
__global__ void compact_kernel(const float* __restrict__ inputs,
                               const float* __restrict__ pos,
                               const float* __restrict__ goal,
                               int* __restrict__ fi, int* __restrict__ fj, float* __restrict__ fcl,
                               int* __restrict__ pi, int* __restrict__ pj,
                               int* __restrict__ gi, int* __restrict__ gj, float* __restrict__ gvl,
                               float* __restrict__ out)
{
    int blk  = blockIdx.x;
    int b    = blk / 3;
    int which= blk % 3;
    int lane = threadIdx.x & 31;

    const float* src; int Kmax; int *oi, *oj; float* ov; float* oidx;
    if (which == 0) {
        src = inputs + ((size_t)b * 2 + 1) * NPIX; Kmax = KF;
        oi = fi + b * KF; oj = fj + b * KF; ov = fcl + b * KF;
        oidx = out + OFF_FRONT + (size_t)b * KF * 2;
    } else if (which == 1) {
        src = pos + (size_t)b * NPIX; Kmax = KP;
        oi = pi + b * KP; oj = pj + b * KP; ov = nullptr;
        oidx = out + OFF_PIDX + (size_t)b * KP * 2;
    } else {
        src = goal + (size_t)b * NPIX; Kmax = KG;
        oi = gi + b * KG; oj = gj + b * KG; ov = gvl + b * KG;
        oidx = out + OFF_GIDX + (size_t)b * KG * 2;
    }

    // fill_value = 0 semantics: pre-zero slots
    for (int s = lane; s < Kmax; s += 32) {
        oi[s] = 0; oj[s] = 0;
        if (ov) ov[s] = 0.0f;
        oidx[2*s] = 0.0f; oidx[2*s+1] = 0.0f;
    }

    unsigned lmask = (1u << lane) - 1u;
    int count = 0;
    for (int base = 0; base < NPIX && count < Kmax; base += 32) {   // 57600 % 32 == 0
        int   idx = base + lane;
        float v   = src[idx];
        bool  nz  = (v != 0.0f);
        unsigned m = __builtin_amdgcn_ballot_w32(nz);
        int slot = count + __popc(m & lmask);
        if (nz && slot < Kmax) {
            int i = idx / HW, j = idx % HW;
            oi[slot] = i; oj[slot] = j;
            if (ov) ov[slot] = v;
            oidx[2*slot]   = (float)i;
            oidx[2*slot+1] = (float)j;
        }
        count += __popc(m);
    }
}

// ---------------------------------------------------------------------------
// Kernel 2: encode() for all 4 point sets.
// Block = 512 threads (16 waves), handles 16 points of one segment.
//  - gather scf (exact, since the bilinear weights are 0/1) into LDS
//  - feat layer0 (256x128 @ 128x16): 16 waves, one 16x16 WMMA tile each
//  - feat layer1 (128x256 @ 256x16): waves 0-7 (WMMA) -> out rows 128..255
//  - loc MLP (5->64->128->128):      waves 8-15 (VALU) -> out rows 0..127
// ---------------------------------------------------------------------------
__global__ void __launch_bounds__(512)
encode_kernel(const float* __restrict__ fm,
              const int*   __restrict__ extras,
              const int* __restrict__ fi, const int* __restrict__ fj, const float* __restrict__ fcl,
              const int* __restrict__ pi, const int* __restrict__ pj,
              const int* __restrict__ gi, const int* __restrict__ gj, const float* __restrict__ gvl,
              const float* __restrict__ fW0, const float* __restrict__ fb0,
              const float* __restrict__ fW1, const float* __restrict__ fb1,
              const float* __restrict__ lW0, const float* __restrict__ lb0,
              const float* __restrict__ lW1, const float* __restrict__ lb1,
              const float* __restrict__ lW2, const float* __restrict__ lb2,
              float* __restrict__ out)
{
    __shared__ float scf[FD * 16];    // [ch][pt]
    __shared__ float hid[256 * 16];   // [m][pt]
    __shared__ float locb[16 * 128];  // per-point h1/h2 buffer
    __shared__ int   s_i[16], s_j[16];
    __shared__ float s_cl[16];

    int blk = blockIdx.x;
    int b   = blk / 57;
    int t   = blk % 57;

    int seg, tile, Kseg; size_t obase; float f3, f4;
    if (t < 32)       { seg = 0; tile = t;      Kseg = KF; obase = OFF_FENC; f3 = 1.0f; f4 = 0.0f; }
    else if (t == 32) { seg = 1; tile = 0;      Kseg = NA; obase = OFF_AENC; f3 = 0.0f; f4 = 1.0f; }
    else if (t < 49)  { seg = 2; tile = t - 33; Kseg = KP; obase = OFF_PENC; f3 = 0.0f; f4 = 1.0f; }
    else              { seg = 3; tile = t - 49; Kseg = KG; obase = OFF_GENC; f3 = 1.0f; f4 = 0.0f; }
    int n0 = tile * 16;

    int tid = threadIdx.x;
    if (tid < 16) {
        int c = n0 + tid;
        int i = 0, j = 0; float cl = 0.0f;
        if (c < Kseg) {
            if (seg == 0)      { i = fi[b*KF+c]; j = fj[b*KF+c]; cl = fcl[b*KF+c]; }
            else if (seg == 1) { i = extras[(b*NA+c)*2+0]; j = extras[(b*NA+c)*2+1]; }
            else if (seg == 2) { i = pi[b*KP+c]; j = pj[b*KP+c]; }
            else               { i = gi[b*KG+c]; j = gj[b*KG+c]; cl = gvl[b*KG+c]; }
        }
        s_i[tid] = i; s_j[tid] = j; s_cl[tid] = cl;
    }
    __syncthreads();

    // gather scf = fm[b, :, i, j]  (exact "interpolation")
    {
        int pt = tid & 15;
        int ii = s_i[pt], jj = s_j[pt];
        for (int it = 0; it < 4; ++it) {
            int ch = (tid >> 4) + it * 32;
            scf[ch * 16 + pt] = fm[(((size_t)b * FD + ch) * HW + ii) * HW + jj];
        }
    }
    __syncthreads();

    int wave = tid >> 5, lane = tid & 31;
    int l = lane & 15, g = lane >> 4;

    // feat layer 0: hid = relu(fW0 @ scf + fb0), 256 rows, 16 waves x 16 rows
    {
        int m0 = wave * 16;
        v8f acc = {};
        for (int k0 = 0; k0 < FD; k0 += 4) {
            int kk = k0 + 2 * g;
            v2f a, bb;
            a.x  = fW0[(m0 + l) * FD + kk];
            a.y  = fW0[(m0 + l) * FD + kk + 1];
            bb.x = scf[kk * 16 + l];
            bb.y = scf[(kk + 1) * 16 + l];
            acc = __builtin_amdgcn_wmma_f32_16x16x4_f32(false, a, false, bb,
                                                        (short)0, acc, false, false);
        }
        for (int r = 0; r < 8; ++r) {
            int m = m0 + r + 8 * g;
            float hv = acc[r] + fb0[m];
            hid[m * 16 + l] = hv > 0.0f ? hv : 0.0f;
        }
    }
    __syncthreads();

    if (wave < 8) {
        // feat layer 1 -> out rows 128..255
        int m0 = wave * 16;
        v8f acc = {};
        for (int k0 = 0; k0 < 256; k0 += 4) {
            int kk = k0 + 2 * g;
            v2f a, bb;
            a.x  = fW1[(m0 + l) * 256 + kk];
            a.y  = fW1[(m0 + l) * 256 + kk + 1];
            bb.x = hid[kk * 16 + l];
            bb.y = hid[(kk + 1) * 16 + l];
            acc = __builtin_amdgcn_wmma_f32_16x16x4_f32(false, a, false, bb,
                                                        (short)0, acc, false, false);
        }
        int c = n0 + l;
        if (c < Kseg) {
            for (int r = 0; r < 8; ++r) {
                int m = m0 + r + 8 * g;
                out[obase + (size_t)(b * 256 + 128 + m) * Kseg + c] = acc[r] + fb1[m];
            }
        }
    } else {
        // loc MLP: 8 waves x 2 points, 32 lanes per point -> out rows 0..127
        for (int pp = 0; pp < 2; ++pp) {
            int p = (wave - 8) * 2 + pp;
            float pts[5];
            pts[0] = ((float)s_i[p] - 120.0f) * (1.0f / 168.0f);
            pts[1] = ((float)s_j[p] - 120.0f) * (1.0f / 168.0f);
            pts[2] = s_cl[p]; pts[3] = f3; pts[4] = f4;
            float* buf = locb + p * 128;
            // h1 (64): 2 per lane
            for (int q = 0; q < 2; ++q) {
                int m = lane * 2 + q;
                float s = lb0[m];
                for (int k = 0; k < 5; ++k) s += lW0[m * 5 + k] * pts[k];
                buf[m] = s > 0.0f ? s : 0.0f;
            }
            // h2 (128): 4 per lane (same-wave LDS ops stay in order)
            float h2[4];
            for (int q = 0; q < 4; ++q) {
                int m = lane * 4 + q;
                float s = lb1[m];
                for (int k = 0; k < 64; ++k) s += lW1[m * 64 + k] * buf[k];
                h2[q] = s > 0.0f ? s : 0.0f;
            }
            for (int q = 0; q < 4; ++q) buf[lane * 4 + q] = h2[q];
            int c = n0 + p;
            if (c < Kseg) {
                for (int q = 0; q < 4; ++q) {
                    int m = lane * 4 + q;
                    float s = lb2[m];
                    for (int k = 0; k < 128; ++k) s += lW2[m * 128 + k] * buf[k];
                    out[obase + (size_t)(b * 256 + m) * Kseg + c] = s;
                }
            }
        }
    }
}

// ---------------------------------------------------------------------------
// Kernel 3: dist MLP. out[b] = dW1(256x128) @ relu(dW0*x + db0)(128x2048) + db1
// One wave per 16x16 output tile; h recomputed on the fly (rank-1 + ReLU).
// 8 * 16 * 128 = 16384 waves.
// ---------------------------------------------------------------------------
__global__ void __launch_bounds__(256)
dist_kernel(const float* __restrict__ dst,
            const int* __restrict__ fi, const int* __restrict__ fj,
            const float* __restrict__ dW0, const float* __restrict__ db0,
            const float* __restrict__ dW1, const float* __restrict__ db1,
            float* __restrict__ out)
{
    int wid  = (int)((blockIdx.x * blockDim.x + threadIdx.x) >> 5);
    int nt   = wid & 127;
    int mt   = (wid >> 7) & 15;
    int b    = wid >> 11;
    int lane = threadIdx.x & 31;
    int l = lane & 15, g = lane >> 4;
    int n0 = nt * 16, m0 = mt * 16;

    // x[c] for column c = n0 + l ; c = d*KF + k -> dist[b, d, fi[k], fj[k]]
    int c = n0 + l;
    int d = c >> 9, k = c & (KF - 1);
    int i = fi[b * KF + k], j = fj[b * KF + k];
    float x  = dst[(((size_t)b * 4 + d) * HW + i) * HW + j];
    float xl = __shfl(x, l, 32);   // broadcast column value to both lane halves

    v8f acc = {};
    for (int k0 = 0; k0 < 128; k0 += 4) {
        int kk = k0 + 2 * g;
        v2f a, bb;
        a.x = dW1[(m0 + l) * 128 + kk];
        a.y = dW1[(m0 + l) * 128 + kk + 1];
        float h0 = dW0[kk]     * xl + db0[kk];
        float h1 = dW0[kk + 1] * xl + db0[kk + 1];
        bb.x = h0 > 0.0f ? h0 : 0.0f;
        bb.y = h1 > 0.0f ? h1 : 0.0f;
        acc = __builtin_amdgcn_wmma_f32_16x16x4_f32(false, a, false, bb,
                                                    (short)0, acc, false, false);
    }
    for (int r = 0; r < 8; ++r) {
        int m = m0 + r + 8 * g;
        out[OFF_DIST + ((size_t)b * 256 + m) * 2048 + n0 + l] = acc[r] + db1[m];
    }
}

// ---------------------------------------------------------------------------
extern "C" void kernel_launch(void* const* d_in, const int* in_sizes, int n_in,
                              void* d_out, int out_size, void* d_ws, size_t ws_size,
                              hipStream_t stream)
{
    const float* fm     = (const float*)d_in[0];
    const float* inputs = (const float*)d_in[1];
    const float* dst    = (const float*)d_in[2];
    const float* pos    = (const float*)d_in[3];
    const float* goal   = (const float*)d_in[4];
    const int*   extras = (const int*)  d_in[5];

    const float *dW0,*db0,*dW1,*db1,*fW0,*fb0,*fW1,*fb1,*lW0,*lb0,*lW1,*lb1,*lW2,*lb2;
    if (in_sizes[6] == 320) {
        // dict insertion order: loc, feat, dist
        lW0=(const float*)d_in[6];  lb0=(const float*)d_in[7];
        lW1=(const float*)d_in[8];  lb1=(const float*)d_in[9];
        lW2=(const float*)d_in[10]; lb2=(const float*)d_in[11];
        fW0=(const float*)d_in[12]; fb0=(const float*)d_in[13];
        fW1=(const float*)d_in[14]; fb1=(const float*)d_in[15];
        dW0=(const float*)d_in[16]; db0=(const float*)d_in[17];
        dW1=(const float*)d_in[18]; db1=(const float*)d_in[19];
    } else {
        // jax tree sorted keys: dist, feat, loc
        dW0=(const float*)d_in[6];  db0=(const float*)d_in[7];
        dW1=(const float*)d_in[8];  db1=(const float*)d_in[9];
        fW0=(const float*)d_in[10]; fb0=(const float*)d_in[11];
        fW1=(const float*)d_in[12]; fb1=(const float*)d_in[13];
        lW0=(const float*)d_in[14]; lb0=(const float*)d_in[15];
        lW1=(const float*)d_in[16]; lb1=(const float*)d_in[17];
        lW2=(const float*)d_in[18]; lb2=(const float*)d_in[19];
    }

    // workspace layout
    int*   fi  = (int*)d_ws;
    int*   fj  = fi + BATCH * KF;
    float* fcl = (float*)(fj + BATCH * KF);
    int*   pi  = (int*)(fcl + BATCH * KF);
    int*   pj  = pi + BATCH * KP;
    int*   gi  = pj + BATCH * KP;
    int*   gj  = gi + BATCH * KG;
    float* gvl = (float*)(gj + BATCH * KG);
    float* out = (float*)d_out;

    compact_kernel<<<24, 32, 0, stream>>>(inputs, pos, goal,
                                          fi, fj, fcl, pi, pj, gi, gj, gvl, out);

    encode_kernel<<<BATCH * 57, 512, 0, stream>>>(fm, extras,
                                                  fi, fj, fcl, pi, pj, gi, gj, gvl,
                                                  fW0, fb0, fW1, fb1,
                                                  lW0, lb0, lW1, lb1, lW2, lb2, out);

    dist_kernel<<<2048, 256, 0, stream>>>(dst, fi, fj, dW0, db0, dW1, db1, out);
}